// GNNPINN_23184233463966
// MI455X (gfx1250) — compile-verified
//
#include <hip/hip_runtime.h>

#define N_NODES 4096
#define E_EDGES 16384
#define HID 128

typedef __attribute__((ext_vector_type(2))) float v2f;
typedef __attribute__((ext_vector_type(8))) float v8f;

// ---------------- degree / inverse-count ----------------
__global__ __launch_bounds__(256) void k_count(const int* __restrict__ dst, float* __restrict__ cnt) {
    int e = blockIdx.x * blockDim.x + threadIdx.x;
    if (e < E_EDGES) atomicAdd(&cnt[dst[e]], 1.0f);
}

__global__ __launch_bounds__(256) void k_inv(float* __restrict__ cnt) {
    int n = blockIdx.x * blockDim.x + threadIdx.x;
    if (n < N_NODES) cnt[n] = 1.0f / fmaxf(cnt[n], 1.0f);
}

// ---------------- layer 1 edge messages (IN=4 -> HID=128) ----------------
__global__ __launch_bounds__(128) void k_msg1(const float* __restrict__ x,   // [N,4]
                                              const float* __restrict__ ea,  // [E,2]
                                              const int* __restrict__ src,
                                              const int* __restrict__ dst,
                                              const float* __restrict__ w1,  // [2, 4*128]
                                              const float* __restrict__ b1,  // [512]
                                              float* __restrict__ agg) {     // [N,128]
    __shared__ float xs[16][4];
    __shared__ float a0s[16], a1s[16];
    __shared__ int ds[16];
    const int t = threadIdx.x;
    const int e0 = blockIdx.x * 16;
    if (t < 16) {
        int e = e0 + t;
        int s = src[e];
        ds[t] = dst[e];
        a0s[t] = ea[e * 2 + 0];
        a1s[t] = ea[e * 2 + 1];
#pragma unroll
        for (int i = 0; i < 4; ++i) xs[t][i] = x[s * 4 + i];
    }
    __syncthreads();
    float w0r[4], w1r[4], br[4];
#pragma unroll
    for (int i = 0; i < 4; ++i) {
        w0r[i] = w1[i * HID + t];
        w1r[i] = w1[512 + i * HID + t];
        br[i]  = b1[i * HID + t];
    }
    for (int e = 0; e < 16; ++e) {
        float a0 = a0s[e], a1 = a1s[e];
        float acc = 0.f;
#pragma unroll
        for (int i = 0; i < 4; ++i) {
            float w = fmaf(a0, w0r[i], fmaf(a1, w1r[i], br[i]));
            w = fmaxf(w, 0.f);
            acc = fmaf(xs[e][i], w, acc);
        }
        atomicAdd(&agg[ds[e] * HID + t], acc);
    }
}

__global__ __launch_bounds__(128) void k_combine1(const float* __restrict__ x,
                                                  const float* __restrict__ agg,
                                                  const float* __restrict__ inv,
                                                  const float* __restrict__ r1, // [4,128]
                                                  const float* __restrict__ c1,
                                                  float* __restrict__ h1) {
    int n = blockIdx.x, o = threadIdx.x;
    float root = 0.f;
#pragma unroll
    for (int i = 0; i < 4; ++i) root = fmaf(x[n * 4 + i], r1[i * HID + o], root);
    float v = fmaf(agg[n * HID + o], inv[n], root + c1[o]);
    h1[n * HID + o] = fmaxf(v, 0.f);
}

// ---------------- layer 2 edge messages (HID -> HID), fused relu-weight-gen + matvec ----------------
__global__ __launch_bounds__(128) void k_msg2(const float* __restrict__ h1,
                                              const float* __restrict__ ea,
                                              const int* __restrict__ src,
                                              const int* __restrict__ dst,
                                              const float* __restrict__ w2, // [2, 128*128]
                                              const float* __restrict__ b2, // [128*128]
                                              float* __restrict__ agg) {
    __shared__ float xs[4][HID];
    __shared__ float a0s[4], a1s[4];
    __shared__ int ds[4];
    const int t = threadIdx.x;
    const int e0 = blockIdx.x * 4;
#pragma unroll
    for (int e = 0; e < 4; ++e) xs[e][t] = h1[src[e0 + e] * HID + t];
    if (t < 4) {
        ds[t] = dst[e0 + t];
        a0s[t] = ea[(e0 + t) * 2 + 0];
        a1s[t] = ea[(e0 + t) * 2 + 1];
    }
    __syncthreads();
    float ra0[4], ra1[4], acc[4];
#pragma unroll
    for (int e = 0; e < 4; ++e) { ra0[e] = a0s[e]; ra1[e] = a1s[e]; acc[e] = 0.f; }
    for (int i = 0; i < HID; ++i) {
        float w0  = w2[i * HID + t];
        float w1v = w2[HID * HID + i * HID + t];
        float bv  = b2[i * HID + t];
#pragma unroll
        for (int e = 0; e < 4; ++e) {
            float w = fmaf(ra0[e], w0, fmaf(ra1[e], w1v, bv));
            w = fmaxf(w, 0.f);
            acc[e] = fmaf(xs[e][i], w, acc[e]);
        }
    }
#pragma unroll
    for (int e = 0; e < 4; ++e) atomicAdd(&agg[ds[e] * HID + t], acc[e]);
}

// ---------------- layer 2 root GEMM: h1[4096,128] @ r2[128,128] via V_WMMA_F32_16X16X4_F32 ----------------
// One wave per 16x16 tile of C; 8 waves/block cover the full 128-wide N for one 16-row M strip.
// A (16x4 f32): lane = (k>>1)*16 + m, vgpr = k&1.  B (4x16): lane = (k>>1)*16 + n, vgpr = k&1.
// C/D (16x16 f32): vgpr r, lanes 0-15 -> M=r, lanes 16-31 -> M=r+8; N = lane&15.
__global__ __launch_bounds__(256) void k_root2_wmma(const float* __restrict__ h1,
                                                    const float* __restrict__ r2, // [128,128] (k,n)
                                                    float* __restrict__ outb) {   // [4096,128]
    const int lane = threadIdx.x & 31;
    const int wave = threadIdx.x >> 5;   // N tile 0..7
    const int m0 = blockIdx.x * 16;
    const int n0 = wave * 16;
    const int r16 = lane & 15;
    const int hi = lane >> 4;            // selects K pair {0,1} vs {2,3}
    v8f acc = {};
    const float* arow = h1 + (m0 + r16) * HID;
    for (int k0 = 0; k0 < HID; k0 += 4) {
        int ka = k0 + hi * 2;
        v2f a, b;
        a.x = arow[ka + 0];
        a.y = arow[ka + 1];
        b.x = r2[(ka + 0) * HID + n0 + r16];
        b.y = r2[(ka + 1) * HID + n0 + r16];
        acc = __builtin_amdgcn_wmma_f32_16x16x4_f32(false, a, false, b, (short)0, acc, false, false);
    }
#pragma unroll
    for (int r = 0; r < 8; ++r)
        outb[(m0 + r + 8 * hi) * HID + n0 + r16] = acc[r];
}

__global__ __launch_bounds__(128) void k_combine2(const float* __restrict__ agg,
                                                  const float* __restrict__ inv,
                                                  const float* __restrict__ rootb,
                                                  const float* __restrict__ c2,
                                                  float* __restrict__ h2) {
    int n = blockIdx.x, o = threadIdx.x;
    float v = fmaf(agg[n * HID + o], inv[n], rootb[n * HID + o] + c2[o]);
    h2[n * HID + o] = fmaxf(v, 0.f);
}

// ---------------- layer 3 edge messages (HID -> 2), one wave32 per edge ----------------
__global__ __launch_bounds__(256) void k_msg3(const float* __restrict__ h2,
                                              const float* __restrict__ ea,
                                              const int* __restrict__ src,
                                              const int* __restrict__ dst,
                                              const float* __restrict__ w3, // [2, 256]
                                              const float* __restrict__ b3, // [256]
                                              float* __restrict__ agg3) {   // [N,2]
    const int lane = threadIdx.x & 31;
    const int wid = threadIdx.x >> 5;
    const int e = blockIdx.x * 8 + wid;
    const int s = src[e], d = dst[e];
    const float a0 = ea[e * 2 + 0], a1 = ea[e * 2 + 1];
    float acc0 = 0.f, acc1 = 0.f;
    for (int i = lane; i < HID; i += 32) {
        float xv = h2[s * HID + i];
        float u0 = fmaxf(fmaf(a0, w3[i * 2 + 0], fmaf(a1, w3[256 + i * 2 + 0], b3[i * 2 + 0])), 0.f);
        float u1 = fmaxf(fmaf(a0, w3[i * 2 + 1], fmaf(a1, w3[256 + i * 2 + 1], b3[i * 2 + 1])), 0.f);
        acc0 = fmaf(xv, u0, acc0);
        acc1 = fmaf(xv, u1, acc1);
    }
    for (int off = 16; off > 0; off >>= 1) {
        acc0 += __shfl_down(acc0, off, 32);
        acc1 += __shfl_down(acc1, off, 32);
    }
    if (lane == 0) {
        atomicAdd(&agg3[d * 2 + 0], acc0);
        atomicAdd(&agg3[d * 2 + 1], acc1);
    }
}

__global__ __launch_bounds__(64) void k_final(const float* __restrict__ h2,
                                              const float* __restrict__ agg3,
                                              const float* __restrict__ inv,
                                              const float* __restrict__ r3, // [128,2]
                                              const float* __restrict__ c3,
                                              float* __restrict__ out) {
    int n = blockIdx.x * blockDim.x + threadIdx.x;
    if (n >= N_NODES) return;
    float acc0 = 0.f, acc1 = 0.f;
    for (int i = 0; i < HID; ++i) {
        float xv = h2[n * HID + i];
        acc0 = fmaf(xv, r3[i * 2 + 0], acc0);
        acc1 = fmaf(xv, r3[i * 2 + 1], acc1);
    }
    float iv = inv[n];
    out[n * 2 + 0] = fmaf(agg3[n * 2 + 0], iv, acc0 + c3[0]);
    out[n * 2 + 1] = fmaf(agg3[n * 2 + 1], iv, acc1 + c3[1]);
}

extern "C" void kernel_launch(void* const* d_in, const int* in_sizes, int n_in,
                              void* d_out, int out_size, void* d_ws, size_t ws_size,
                              hipStream_t stream) {
    const float* x  = (const float*)d_in[0];
    const float* ea = (const float*)d_in[1];
    const int*   src = (const int*)d_in[2];
    const int*   dst = (const int*)d_in[3];
    const float* w1 = (const float*)d_in[4];
    const float* b1 = (const float*)d_in[5];
    const float* r1 = (const float*)d_in[6];
    const float* c1 = (const float*)d_in[7];
    const float* w2 = (const float*)d_in[8];
    const float* b2 = (const float*)d_in[9];
    const float* r2 = (const float*)d_in[10];
    const float* c2 = (const float*)d_in[11];
    const float* w3 = (const float*)d_in[12];
    const float* b3 = (const float*)d_in[13];
    const float* r3 = (const float*)d_in[14];
    const float* c3 = (const float*)d_in[15];
    float* out = (float*)d_out;

    float* ws    = (float*)d_ws;
    float* h1    = ws;                      // 4096*128
    float* h2    = h1 + N_NODES * HID;      // 4096*128
    float* agg   = h2 + N_NODES * HID;      // 4096*128 (reused per layer)
    float* rootb = agg + N_NODES * HID;     // 4096*128
    float* inv   = rootb + N_NODES * HID;   // 4096

    // degree -> 1/max(deg,1) (shared across layers; same dst)
    hipMemsetAsync(inv, 0, N_NODES * sizeof(float), stream);
    k_count<<<E_EDGES / 256, 256, 0, stream>>>(dst, inv);
    k_inv<<<N_NODES / 256, 256, 0, stream>>>(inv);

    // layer 1
    hipMemsetAsync(agg, 0, N_NODES * HID * sizeof(float), stream);
    k_msg1<<<E_EDGES / 16, 128, 0, stream>>>(x, ea, src, dst, w1, b1, agg);
    k_combine1<<<N_NODES, 128, 0, stream>>>(x, agg, inv, r1, c1, h1);

    // layer 2
    hipMemsetAsync(agg, 0, N_NODES * HID * sizeof(float), stream);
    k_msg2<<<E_EDGES / 4, 128, 0, stream>>>(h1, ea, src, dst, w2, b2, agg);
    k_root2_wmma<<<N_NODES / 16, 256, 0, stream>>>(h1, r2, rootb);
    k_combine2<<<N_NODES, 128, 0, stream>>>(agg, inv, rootb, c2, h2);

    // layer 3
    hipMemsetAsync(agg, 0, N_NODES * 2 * sizeof(float), stream);
    k_msg3<<<E_EDGES / 8, 256, 0, stream>>>(h2, ea, src, dst, w3, b3, agg);
    k_final<<<(N_NODES + 63) / 64, 64, 0, stream>>>(h2, agg, inv, r3, c3, out);
}